// LateralInhibition_67327907332794
// MI455X (gfx1250) — compile-verified
//
#include <hip/hip_runtime.h>
#include <math.h>

// ---------------- problem constants ----------------
#define LK      7
#define NTAPS   49
#define IMG     512
#define NPIXIMG (IMG * IMG)          // 262144
#define NIMG    16
#define NPIX    ((size_t)NIMG * NPIXIMG)  // 4194304
#define A_COEF  0.1f
#define B_COEF  0.9f

// tile geometry for pass1: each block computes a 16(x) x 8(y) output tile,
// one wave32 per output row (16 pixels per wave, WMMA 16x16x4 f32).
#define TW       16
#define TH       8
#define HALO     3
#define TILE_W   (TW + 2 * HALO)     // 22
#define TILE_HA  15                  // 14 needed + 1 guard row (finite pad reads)
#define TILE_S   24                  // padded row stride (floats)

typedef __attribute__((ext_vector_type(2))) float v2f;
typedef __attribute__((ext_vector_type(8))) float v8f;

// ---- gfx1250 async global->LDS path (guarded; falls back to plain loads) ----
#if defined(__has_builtin)
#  if __has_builtin(__builtin_amdgcn_global_load_async_to_lds_b32) && \
      __has_builtin(__builtin_amdgcn_s_wait_asynccnt)
#    define USE_ASYNC_LDS 1
#  endif
#endif
#ifndef USE_ASYNC_LDS
#  define USE_ASYNC_LDS 0
#endif

#if USE_ASYNC_LDS
typedef __attribute__((address_space(1))) int g_i32_t;   // global int
typedef __attribute__((address_space(3))) int l_i32_t;   // LDS int
#endif

// ---------------------------------------------------------------------------
// pass1: per-pixel average_term = exp(-mean(patch)) and
//        differential_term = sum_k w[k] * relu(patch[k] - center)
//
// WMMA mapping (V_WMMA_F32_16X16X4_F32, 14 K-chunks):
//   chunk i carries taps {2i, 2i+1} (lanes 0-15, K=0,1) and
//   taps {28+2i, 28+2i+1} (lanes 16-31, K=2,3).
//   Since 28 = 4*7, pos(t+28) = pos(t) + 4 rows -> upper lanes use the same
//   compile-time LDS offsets from a base shifted by 96 floats. All tap/weight
//   loads collapse to ds_load_2addr with lane-uniform immediate offsets.
//   Taps 49..55 are padding: weight 0, A values finite (real tile data).
// ---------------------------------------------------------------------------
__global__ __launch_bounds__(256) void li_pass1(const float* __restrict__ x,
                                                const float* __restrict__ kern,
                                                float* __restrict__ out_avg,
                                                float* __restrict__ out_diff) {
  __shared__ float tile[TILE_HA * TILE_S];  // 15 x 24 floats
  __shared__ float kw[56];                  // 49 weights, zero padded to 56
  __shared__ float diffbuf[TH][TW];

  const int tid = threadIdx.x;
  const int bx  = blockIdx.x;              // 0..31  (x tiles of 16)
  const int by  = blockIdx.y;              // 0..63  (y tiles of 8)
  const int img = blockIdx.z;              // 0..15
  const float* xi = x + (size_t)img * NPIXIMG;

  if (tid < 56) kw[tid] = (tid < NTAPS) ? kern[tid] : 0.0f;

  // ---- stage 22x15 halo tile into LDS (zero outside the image) ----
  const int y0 = by * TH - HALO;
  const int x0 = bx * TW - HALO;
  for (int i = tid; i < TILE_HA * TILE_W; i += 256) {
    const int r = i / TILE_W;
    const int c = i - r * TILE_W;
    const int gy = y0 + r;
    const int gx = x0 + c;
    float* lp = &tile[r * TILE_S + c];
    if ((unsigned)gy < (unsigned)IMG && (unsigned)gx < (unsigned)IMG) {
#if USE_ASYNC_LDS
      __builtin_amdgcn_global_load_async_to_lds_b32(
          (g_i32_t*)(xi + gy * IMG + gx), (l_i32_t*)lp, 0, 0);
#else
      *lp = xi[gy * IMG + gx];
#endif
    } else {
      *lp = 0.0f;
    }
  }
#if USE_ASYNC_LDS
  __builtin_amdgcn_s_wait_asynccnt(0);
#endif
  __syncthreads();

  // ---- WMMA loop: one wave per output row, 16 pixels per wave ----
  const int wv   = tid >> 5;   // wave id 0..7 -> output row within tile
  const int lane = tid & 31;
  const int hf   = lane >> 4;  // 0: taps 0..27 (K0,K1); 1: taps 28..55 (K2,K3)
  const int n    = lane & 15;  // pixel (M for A, N for B/C/D)

  // per-lane bases; upper half shifted by pos(28) = 4 rows = 96 floats
  const float* tbase = &tile[wv * TILE_S + n + hf * (4 * TILE_S)];
  const float* kwb   = &kw[hf * 28];
  const float  center = tile[(wv + HALO) * TILE_S + (n + HALO)];

  v8f   c   = {};    // f32 16x16 accumulator (every N column identical)
  float raw = 0.0f;  // raw patch sum (this lane-half's real taps)

#pragma unroll
  for (int i = 0; i < 14; ++i) {
    const int t0 = 2 * i;
    const int t1 = 2 * i + 1;
    const int o0 = (t0 / LK) * TILE_S + (t0 % LK);  // compile-time constants
    const int o1 = (t1 / LK) * TILE_S + (t1 % LK);
    const float p0 = tbase[o0];
    const float p1 = tbase[o1];
    const float w0 = kwb[t0];   // zero for padded taps (t+28 >= 49)
    const float w1 = kwb[t1];

    // raw sum of real taps only (lower half: all real; upper: 28+t <= 48)
    if (i <= 9)        raw += p0 + p1;
    else if (i == 10)  raw += p0 + (hf ? 0.0f : p1);   // tap 48 real, 49 pad
    else               raw += hf ? 0.0f : (p0 + p1);   // taps 50..55 pad

    v2f a, b;
    a.x = fmaxf(p0 - center, 0.0f);
    a.y = fmaxf(p1 - center, 0.0f);
    b.x = w0;
    b.y = w1;
    // D = A(16x4 relu) x B(4x16 broadcast weights) + C
    c = __builtin_amdgcn_wmma_f32_16x16x4_f32(false, a, false, b,
                                              (short)0, c, false, false);
  }

  // combine the two lane-halves' disjoint tap sums -> full 49-tap sum
  const float tot = raw + __shfl_xor(raw, 16, 32);

  // D layout: VGPR g -> lanes 0-15 hold D[g, n] (all n equal), lanes 16-31
  // hold D[g+8, n]. Lane 0 / lane 16 own diff[0..7] / diff[8..15].
  if ((lane & 15) == 0) {
#pragma unroll
    for (int j = 0; j < 8; ++j) diffbuf[wv][hf * 8 + j] = c[j];
  }
  __syncthreads();

  if (lane < 16) {
    const int row = by * TH + wv;
    const int col = bx * TW + n;
    const size_t idx = (size_t)img * NPIXIMG + (size_t)row * IMG + col;
    out_avg[idx]  = expf(-tot * (1.0f / 49.0f));
    out_diff[idx] = diffbuf[wv][n];
  }
}

// ---------------------------------------------------------------------------
// deterministic two-level reduction of sum((A*avg + B*diff)^2)
// ---------------------------------------------------------------------------
__global__ __launch_bounds__(256) void li_reduce1(const float* __restrict__ avg,
                                                  const float* __restrict__ dif,
                                                  float* __restrict__ partial) {
  __shared__ float sm[256];
  const int tid = threadIdx.x;
  const size_t base = (size_t)blockIdx.x * 4096;
  float acc = 0.0f;
  for (int j = tid; j < 4096; j += 256) {
    const float m = fmaf(B_COEF, dif[base + j], A_COEF * avg[base + j]);
    acc = fmaf(m, m, acc);
  }
  sm[tid] = acc;
  __syncthreads();
  for (int s = 128; s > 0; s >>= 1) {
    if (tid < s) sm[tid] += sm[tid + s];
    __syncthreads();
  }
  if (tid == 0) partial[blockIdx.x] = sm[0];
}

__global__ __launch_bounds__(256) void li_reduce2(const float* __restrict__ partial,
                                                  float* __restrict__ inv_norm) {
  __shared__ float sm[256];
  const int tid = threadIdx.x;
  float acc = 0.0f;
  for (int j = tid; j < 1024; j += 256) acc += partial[j];
  sm[tid] = acc;
  __syncthreads();
  for (int s = 128; s > 0; s >>= 1) {
    if (tid < s) sm[tid] += sm[tid + s];
    __syncthreads();
  }
  if (tid == 0) inv_norm[0] = 1.0f / sqrtf(sm[0]);
}

// ---------------------------------------------------------------------------
// pass3: suppression = where(x > mask * inv_norm, x, 0)   (float4 vectorized)
// ---------------------------------------------------------------------------
__global__ __launch_bounds__(256) void li_pass3(const float* __restrict__ x,
                                                const float* __restrict__ avg,
                                                const float* __restrict__ dif,
                                                const float* __restrict__ inv_norm,
                                                float* __restrict__ sup) {
  const size_t i4 = ((size_t)blockIdx.x * 256 + threadIdx.x) * 4;
  const float4 xv = *(const float4*)(x + i4);
  const float4 av = *(const float4*)(avg + i4);
  const float4 dv = *(const float4*)(dif + i4);
  const float inv = inv_norm[0];
  float4 s;
  s.x = (xv.x > fmaf(B_COEF, dv.x, A_COEF * av.x) * inv) ? xv.x : 0.0f;
  s.y = (xv.y > fmaf(B_COEF, dv.y, A_COEF * av.y) * inv) ? xv.y : 0.0f;
  s.z = (xv.z > fmaf(B_COEF, dv.z, A_COEF * av.z) * inv) ? xv.z : 0.0f;
  s.w = (xv.w > fmaf(B_COEF, dv.w, A_COEF * av.w) * inv) ? xv.w : 0.0f;
  *(float4*)(sup + i4) = s;
}

// ---------------------------------------------------------------------------
extern "C" void kernel_launch(void* const* d_in, const int* in_sizes, int n_in,
                              void* d_out, int out_size, void* d_ws, size_t ws_size,
                              hipStream_t stream) {
  const float* x    = (const float*)d_in[0];  // [16,1,512,512]
  const float* kern = (const float*)d_in[1];  // [49]
  float* out = (float*)d_out;
  float* sup = out;                 // output 0: suppression
  float* avg = out + NPIX;          // output 1: average_term
  float* dif = out + 2 * NPIX;      // output 2: differential_term

  float* partial  = (float*)d_ws;   // 1024 floats
  float* inv_norm = partial + 1024; // 1 float

  dim3 g1(IMG / TW, IMG / TH, NIMG);            // 32 x 64 x 16
  li_pass1<<<g1, 256, 0, stream>>>(x, kern, avg, dif);
  li_reduce1<<<1024, 256, 0, stream>>>(avg, dif, partial);
  li_reduce2<<<1, 256, 0, stream>>>(partial, inv_norm);
  li_pass3<<<(unsigned)(NPIX / 1024), 256, 0, stream>>>(x, avg, dif, inv_norm, sup);
}